// SinenetLayerTooBig_76433238000020
// MI455X (gfx1250) — compile-verified
//
#include <hip/hip_runtime.h>
#include <math.h>

// Problem constants (from the reference)
#define S_DIM   16
#define B_DIM   64
#define SB      (S_DIM * B_DIM)      // 1024
#define T_DIM   640
#define D_DIM   256
#define NUM_CH  4
#define NF      (D_DIM / NUM_CH)     // 64 harmonics
#define M_DIM   (2 * NF)             // 128 = [sin rows | cos rows]

#define LOG_F_MEAN 5.02654f
#define LOG_F_STD  0.373288f

typedef float v2f __attribute__((ext_vector_type(2)));
typedef float v8f __attribute__((ext_vector_type(8)));

// ---------------------------------------------------------------------------
// Phase 0: build projection matrix M (M_DIM x D_DIM), data-independent.
//   M[d>>2      , d] = a[d] * cos(phi[d])
//   M[64+(d>>2) , d] = a[d] * sin(phi[d])
//   all other entries zero.
// ---------------------------------------------------------------------------
__global__ void build_M_kernel(const float* __restrict__ a,
                               const float* __restrict__ phi,
                               float* __restrict__ Mm) {
    int idx = blockIdx.x * blockDim.x + threadIdx.x;   // 0 .. 128*256-1
    int m = idx >> 8;          // row 0..127
    int d = idx & 255;         // col 0..255
    int n = d >> 2;            // harmonic row index (0-based)
    float v = 0.0f;
    if (m == n) {
        v = a[d] * cosf(phi[d]);
    } else if (m == n + NF) {
        v = a[d] * sinf(phi[d]);
    }
    Mm[m * D_DIM + d] = v;
}

// ---------------------------------------------------------------------------
// Phase 1: per batch element sb, compute
//   SC[sb, m]      = sum_t sin((m+1)    * theta_t) * x[sb,t]   (m <  64)
//   SC[sb, m]      = sum_t cos((m-63)   * theta_t) * x[sb,t]   (m >= 64)
// where theta_t (in *revolutions*) = f * (k_t - tau),  f = exp(nlf*std+mean).
// One 128-thread block per sb; x and theta staged in LDS.
// Waves 0-1 -> sin, waves 2-3 -> cos (wave-uniform, no divergence).
// v_sin_f32 / v_cos_f32 compute sin/cos(2*pi * input), input range-reduced
// to [-0.5, 0.5] revolutions with rintf (v_rndne_f32).
// ---------------------------------------------------------------------------
__global__ void __launch_bounds__(128)
phase1_kernel(const float* __restrict__ x,
              const float* __restrict__ nlf,
              const float* __restrict__ tau,
              const float* __restrict__ kT,
              float* __restrict__ SC) {
    __shared__ float th_s[T_DIM];
    __shared__ float x_s[T_DIM];

    const int sb  = blockIdx.x;      // 0..1023
    const int tid = threadIdx.x;     // 0..127

    const float f  = expf(nlf[sb] * LOG_F_STD + LOG_F_MEAN);
    const float tv = tau[sb];

    // Stage x[sb, :] and theta (revolutions of harmonic 1) into LDS.
    for (int t = tid; t < T_DIM; t += 128) {
        x_s[t]  = x[sb * T_DIM + t];
        th_s[t] = f * (kT[t] - tv);
    }
    __syncthreads();

    const float nf     = (float)((tid & (NF - 1)) + 1);  // harmonic 1..64
    const bool  do_cos = (tid >= NF);                    // wave-uniform

    float acc = 0.0f;
    #pragma unroll 8
    for (int t = 0; t < T_DIM; ++t) {
        float arg = nf * th_s[t];            // revolutions
        float r   = arg - rintf(arg);        // [-0.5, 0.5]
        float sv  = do_cos ? __builtin_amdgcn_cosf(r)
                           : __builtin_amdgcn_sinf(r);
        acc = fmaf(sv, x_s[t], acc);
    }

    SC[sb * M_DIM + tid] = acc;
}

// ---------------------------------------------------------------------------
// Phase 2: out[1024 x 256] = SC[1024 x 128] @ M[128 x 256] in f32 via
// V_WMMA_F32_16X16X4_F32. One 16(sb) x 16(d) tile per wave, K-loop of 32.
//
// A (16x4 f32) layout: lanes 0-15 hold rows M=0..15 with K={0,1} in the two
// VGPRs; lanes 16-31 hold the same rows with K={2,3}.
// B (4x16 f32) layout: lanes 0-15 hold cols N=0..15 of rows K={0,1};
// lanes 16-31 hold rows K={2,3}.
// C/D (16x16 f32, 8 VGPRs): VGPR r -> row r (lanes 0-15) / row r+8
// (lanes 16-31), N = lane & 15.
// ---------------------------------------------------------------------------
__global__ void __launch_bounds__(128)
phase2_wmma_kernel(const float* __restrict__ SC,
                   const float* __restrict__ Mm,
                   float* __restrict__ out) {
    const int lane = threadIdx.x & 31;
    const int wave = threadIdx.x >> 5;
    const int tile = blockIdx.x * 4 + wave;   // 0..1023 tiles
    const int trow = tile >> 4;               // sb-tile  0..63
    const int tcol = tile & 15;               // d-tile   0..15

    const int r16   = lane & 15;              // row (A) / col (B) within tile
    const int khalf = (lane >> 4) << 1;       // 0 for lanes 0-15, 2 for 16-31

    const float* Abase = SC + (size_t)(trow * 16 + r16) * M_DIM;
    const float* Bbase = Mm + (size_t)(tcol * 16 + r16);

    v8f c = {};
    #pragma unroll
    for (int k0 = 0; k0 < M_DIM; k0 += 4) {
        v2f a, b;
        a.x = Abase[k0 + khalf];
        a.y = Abase[k0 + khalf + 1];
        b.x = Bbase[(size_t)(k0 + khalf) * D_DIM];
        b.y = Bbase[(size_t)(k0 + khalf + 1) * D_DIM];
        // (neg_a, A, neg_b, B, c_mod, C, reuse_a, reuse_b)
        c = __builtin_amdgcn_wmma_f32_16x16x4_f32(
                false, a, false, b, (short)0, c, false, false);
    }

    const int rowbase = trow * 16 + ((lane >> 4) << 3);  // +0 or +8
    const int col     = tcol * 16 + (lane & 15);
    #pragma unroll
    for (int r = 0; r < 8; ++r) {
        out[(size_t)(rowbase + r) * D_DIM + col] = c[r];
    }
}

// ---------------------------------------------------------------------------
// Launch: inputs in setup_inputs() order:
//   0: x (S,B,1,T)  1: nlf (S,B,1,1)  2: tau (S,B,1,1)
//   3: a (D,1)      4: phi (D,1)      5: i_2pi (D,1)   6: k_T (1,T)
// out: (S,B,D) f32.  Workspace: SC (1024*128 f32) + M (128*256 f32).
// ---------------------------------------------------------------------------
extern "C" void kernel_launch(void* const* d_in, const int* in_sizes, int n_in,
                              void* d_out, int out_size, void* d_ws, size_t ws_size,
                              hipStream_t stream) {
    (void)in_sizes; (void)n_in; (void)out_size; (void)ws_size;

    const float* x   = (const float*)d_in[0];
    const float* nlf = (const float*)d_in[1];
    const float* tau = (const float*)d_in[2];
    const float* a   = (const float*)d_in[3];
    const float* phi = (const float*)d_in[4];
    // d_in[5] (i_2pi) is implied by the d-index; d_in[6] is k_T.
    const float* kT  = (const float*)d_in[6];

    float* SC  = (float*)d_ws;                       // 1024 * 128 f32
    float* Mm  = SC + (size_t)SB * M_DIM;            // 128  * 256 f32
    float* out = (float*)d_out;

    // Phase 0: build projection matrix (128*256 elements).
    build_M_kernel<<<(M_DIM * D_DIM) / 256, 256, 0, stream>>>(a, phi, Mm);

    // Phase 1: harmonic sin/cos correlations -> SC.
    phase1_kernel<<<SB, 128, 0, stream>>>(x, nlf, tau, kT, SC);

    // Phase 2: out = SC @ M via f32 WMMA. 1024 tiles, 4 waves/block.
    phase2_wmma_kernel<<<SB * D_DIM / (16 * 16) / 4, 128, 0, stream>>>(SC, Mm, out);
}